// GNNLandmarkSelector_22179211117042
// MI455X (gfx1250) — compile-verified
//
#include <hip/hip_runtime.h>
#include <hip/hip_bf16.h>
#include <math.h>

typedef __attribute__((ext_vector_type(2))) float v2f;
typedef __attribute__((ext_vector_type(8))) float v8f;

// ---------------------------------------------------------------------------
// init: acc1[N*16]=0, acc2[N*8]=0, deg[N]=1 (self-loop)
// ---------------------------------------------------------------------------
__global__ void gcn_init(float* __restrict__ acc1, float* __restrict__ acc2,
                         float* __restrict__ deg, int n) {
  int i = blockIdx.x * blockDim.x + threadIdx.x;
  if (i < n * 16) acc1[i] = 0.0f;
  if (i < n * 8)  acc2[i] = 0.0f;
  if (i < n)      deg[i]  = 1.0f;
}

// ---------------------------------------------------------------------------
// degree count over dst
// ---------------------------------------------------------------------------
__global__ void gcn_deg(const int* __restrict__ dst, float* __restrict__ deg, int nE) {
  int e = blockIdx.x * blockDim.x + threadIdx.x;
  if (e < nE) atomicAdd(&deg[dst[e]], 1.0f);
}

__global__ void gcn_dinv(const float* __restrict__ deg, float* __restrict__ dinv, int n) {
  int i = blockIdx.x * blockDim.x + threadIdx.x;
  if (i < n) dinv[i] = rsqrtf(deg[i]);   // deg >= 1 always (self-loop)
}

// ---------------------------------------------------------------------------
// layer-1 matmul via V_WMMA_F32_16X16X4_F32: g1 = (x[Nx6] @ W1[6x16]) * dinv
// one wave handles a 16-row tile; K=6 -> two WMMAs (K padded to 8 with zeros)
// A layout: lanes 0-15 hold row M=l, K={k0,k0+1}; lanes 16-31 K={k0+2,k0+3}
// ---------------------------------------------------------------------------
__global__ void __launch_bounds__(128)
gcn_mm1_wmma(const float* __restrict__ x, const float* __restrict__ W,
             const float* __restrict__ dinv, float* __restrict__ g, int n) {
  const int lane = threadIdx.x & 31;
  const int wave = blockIdx.x * 4 + (threadIdx.x >> 5);
  const int base = wave * 16;
  if (base >= n) return;                 // wave-uniform: EXEC all-1 when active
  const int l  = lane & 15;
  const int hi = lane >> 4;
  const int row = base + l;

  const float* xr = x + (long)row * 6;
  v2f a0, a1, b0, b1;
  // step 0: K = {0,1} (lanes 0-15) / {2,3} (lanes 16-31)
  a0.x = xr[2 * hi + 0];
  a0.y = xr[2 * hi + 1];
  b0.x = W[(2 * hi + 0) * 16 + l];
  b0.y = W[(2 * hi + 1) * 16 + l];
  // step 1: K = {4,5} / {6,7}=zero-pad
  a1.x = hi ? 0.0f : xr[4];
  a1.y = hi ? 0.0f : xr[5];
  b1.x = hi ? 0.0f : W[4 * 16 + l];
  b1.y = hi ? 0.0f : W[5 * 16 + l];

  v8f c = {};
  c = __builtin_amdgcn_wmma_f32_16x16x4_f32(false, a0, false, b0, (short)0, c, false, false);
  c = __builtin_amdgcn_wmma_f32_16x16x4_f32(false, a1, false, b1, (short)0, c, false, false);

  // D: lanes 0-15 -> M=j, lanes 16-31 -> M=j+8; N = l. Fold dinv[src] in here.
#pragma unroll
  for (int j = 0; j < 8; ++j) {
    int r = base + j + 8 * hi;
    g[(long)r * 16 + l] = c[j] * dinv[r];
  }
}

// ---------------------------------------------------------------------------
// layer-2 matmul: g2 = (h1[Nx16] @ W2[16x8]) * dinv  (N padded 8->16)
// ---------------------------------------------------------------------------
__global__ void __launch_bounds__(128)
gcn_mm2_wmma(const float* __restrict__ h, const float* __restrict__ W,
             const float* __restrict__ dinv, float* __restrict__ g, int n) {
  const int lane = threadIdx.x & 31;
  const int wave = blockIdx.x * 4 + (threadIdx.x >> 5);
  const int base = wave * 16;
  if (base >= n) return;
  const int l  = lane & 15;
  const int hi = lane >> 4;
  const int row = base + l;

  const float* hr = h + (long)row * 16;
  const bool keep = (l < 8);

  v8f c = {};
#pragma unroll
  for (int k0 = 0; k0 < 16; k0 += 4) {
    const int k = k0 + 2 * hi;
    v2f a, b;
    a.x = hr[k];
    a.y = hr[k + 1];
    b.x = keep ? W[k * 8 + l] : 0.0f;        // zero-pad B columns 8..15
    b.y = keep ? W[(k + 1) * 8 + l] : 0.0f;
    c = __builtin_amdgcn_wmma_f32_16x16x4_f32(false, a, false, b, (short)0, c, false, false);
  }

  if (keep) {
#pragma unroll
    for (int j = 0; j < 8; ++j) {
      int r = base + j + 8 * hi;
      g[(long)r * 8 + l] = c[j] * dinv[r];
    }
  }
}

// ---------------------------------------------------------------------------
// edge scatter: acc[dst][f] += g[src][f]; one thread per (edge, feature)
// ---------------------------------------------------------------------------
__global__ void gcn_scatter16(const int* __restrict__ src, const int* __restrict__ dst,
                              const float* __restrict__ g, float* __restrict__ acc,
                              long long total) {
  long long idx = (long long)blockIdx.x * blockDim.x + threadIdx.x;
  if (idx >= total) return;
  int e = (int)(idx >> 4);
  int f = (int)(idx & 15);
  int s = src[e], d = dst[e];
  atomicAdd(&acc[(long)d * 16 + f], g[(long)s * 16 + f]);
}

__global__ void gcn_scatter8(const int* __restrict__ src, const int* __restrict__ dst,
                             const float* __restrict__ g, float* __restrict__ acc,
                             long long total) {
  long long idx = (long long)blockIdx.x * blockDim.x + threadIdx.x;
  if (idx >= total) return;
  int e = (int)(idx >> 3);
  int f = (int)(idx & 7);
  int s = src[e], d = dst[e];
  atomicAdd(&acc[(long)d * 8 + f], g[(long)s * 8 + f]);
}

// ---------------------------------------------------------------------------
// layer-1 finalize: h1 = relu(dinv*(acc1 + g1) + b1)
// ---------------------------------------------------------------------------
__global__ void gcn_fin1(const float* __restrict__ acc, const float* __restrict__ g,
                         const float* __restrict__ dinv, const float* __restrict__ b,
                         float* __restrict__ h, int n) {
  int idx = blockIdx.x * blockDim.x + threadIdx.x;
  if (idx >= n * 16) return;
  int node = idx >> 4;
  int f = idx & 15;
  float v = dinv[node] * (acc[idx] + g[idx]) + b[f];
  h[idx] = v > 0.0f ? v : 0.0f;
}

// ---------------------------------------------------------------------------
// layer-2 finalize fused with FC + sigmoid:
// out[n] = sigmoid( sum_f relu(dinv*(acc2+g2)+b2)[f] * Wfc[f] + bfc )
// ---------------------------------------------------------------------------
__global__ void gcn_fin2(const float* __restrict__ acc, const float* __restrict__ g,
                         const float* __restrict__ dinv, const float* __restrict__ b,
                         const float* __restrict__ Wfc, const float* __restrict__ bfc,
                         float* __restrict__ out, int n) {
  int node = blockIdx.x * blockDim.x + threadIdx.x;
  if (node >= n) return;
  float di = dinv[node];
  float s = bfc[0];
#pragma unroll
  for (int f = 0; f < 8; ++f) {
    float v = di * (acc[(long)node * 8 + f] + g[(long)node * 8 + f]) + b[f];
    v = v > 0.0f ? v : 0.0f;
    s += v * Wfc[f];
  }
  out[node] = 1.0f / (1.0f + expf(-s));
}

// ---------------------------------------------------------------------------
extern "C" void kernel_launch(void* const* d_in, const int* in_sizes, int n_in,
                              void* d_out, int out_size, void* d_ws, size_t ws_size,
                              hipStream_t stream) {
  const float* x   = (const float*)d_in[0];   // [N,6]
  const int*   ei  = (const int*)d_in[1];     // [2,E]
  const float* W1  = (const float*)d_in[2];   // [6,16]
  const float* b1  = (const float*)d_in[3];   // [16]
  const float* W2  = (const float*)d_in[4];   // [16,8]
  const float* b2  = (const float*)d_in[5];   // [8]
  const float* Wfc = (const float*)d_in[6];   // [8]
  const float* bfc = (const float*)d_in[7];   // [1]
  float* out = (float*)d_out;

  const int N = in_sizes[0] / 6;              // 200,000 (multiple of 16)
  const int E = in_sizes[1] / 2;              // 6,400,000
  const int* src = ei;
  const int* dst = ei + E;

  // workspace layout (floats)
  float* ws   = (float*)d_ws;
  float* deg  = ws;                 // N
  float* dinv = deg  + N;           // N
  float* g1   = dinv + N;           // N*16
  float* acc1 = g1   + (long)N*16;  // N*16
  float* h1   = acc1 + (long)N*16;  // N*16
  float* g2   = h1   + (long)N*16;  // N*8
  float* acc2 = g2   + (long)N*8;   // N*8

  const int TB = 256;
  const long long tot16 = (long long)E * 16;
  const long long tot8  = (long long)E * 8;

  gcn_init<<<(N * 16 + TB - 1) / TB, TB, 0, stream>>>(acc1, acc2, deg, N);
  gcn_deg<<<(E + TB - 1) / TB, TB, 0, stream>>>(dst, deg, E);
  gcn_dinv<<<(N + TB - 1) / TB, TB, 0, stream>>>(deg, dinv, N);

  // layer 1
  {
    int waves = (N + 15) / 16;                    // 12,500
    int blocks = (waves + 3) / 4;                 // 4 waves (128 threads) per block
    gcn_mm1_wmma<<<blocks, 128, 0, stream>>>(x, W1, dinv, g1, N);
  }
  gcn_scatter16<<<(int)((tot16 + TB - 1) / TB), TB, 0, stream>>>(src, dst, g1, acc1, tot16);
  gcn_fin1<<<(N * 16 + TB - 1) / TB, TB, 0, stream>>>(acc1, g1, dinv, b1, h1, N);

  // layer 2
  {
    int waves = (N + 15) / 16;
    int blocks = (waves + 3) / 4;
    gcn_mm2_wmma<<<blocks, 128, 0, stream>>>(h1, W2, dinv, g2, N);
  }
  gcn_scatter8<<<(int)((tot8 + TB - 1) / TB), TB, 0, stream>>>(src, dst, g2, acc2, tot8);
  gcn_fin2<<<(N + TB - 1) / TB, TB, 0, stream>>>(acc2, g2, dinv, b2, Wfc, bfc, out, N);
}